// FLAME_84585085927673
// MI455X (gfx1250) — compile-verified
//
#include <hip/hip_runtime.h>

// ---------------- problem constants ----------------
#define B_    1024
#define V_    5023
#define N_    15069          // V*3
#define NP_   15104          // N padded to 118*128
#define L_    150            // shape+expr params
#define KP_   152            // L padded to mult of 4 (and 8)
#define KPP_  40             // 36 pose features padded
#define J_    5
#define FC_   9976
#define NLMK_ 68

typedef float v2f __attribute__((ext_vector_type(2)));
typedef float v8f __attribute__((ext_vector_type(8)));
typedef unsigned int u32;
typedef u32 v4u_t __attribute__((ext_vector_type(4)));
typedef int  v8i_t __attribute__((ext_vector_type(8)));
typedef int  v4i_t __attribute__((ext_vector_type(4)));

// ---------------- TDM: 2D tile (rows x width f32, row stride == width) -> LDS
// D# layout per CDNA5 ISA (group0 128b, group1 256b).
// This toolchain exposes the 6-arg builtin: (g0, g1, g2, g3, g4, cpol).
__device__ __forceinline__ void tdm_load_2d(u32 lds_addr, const void* gptr,
                                            u32 width, u32 rows) {
  unsigned long long ga = (unsigned long long)gptr;
  v4u_t g0;
  g0.x = 1u;                                            // count=1, user desc
  g0.y = lds_addr;                                      // lds_addr[31:0]
  g0.z = (u32)(ga & 0xffffffffull);                     // global_addr[31:0]
  g0.w = (u32)((ga >> 32) & 0x1ffffffull) | (2u << 30); // ga[56:32] | type=2
  v8i_t g1;
  g1[0] = (int)(2u << 16);                              // data_size=2 -> 4B
  g1[1] = (int)((width & 0xffffu) << 16);               // tensor_dim0[15:0]
  g1[2] = (int)((width >> 16) | ((rows & 0xffffu) << 16)); // td0 hi | td1 lo
  g1[3] = (int)((rows >> 16) | ((width & 0xffffu) << 16)); // td1 hi | tile_dim0
  g1[4] = (int)(rows & 0xffffu);                        // tile_dim1 | tile_dim2=0
  g1[5] = (int)width;                                   // tensor_dim0_stride lo32
  g1[6] = 0;                                            // stride0 hi | stride1 lo
  g1[7] = 0;
  v4i_t z4 = (v4i_t)0;
  v8i_t z8 = (v8i_t)0;
  __builtin_amdgcn_tensor_load_to_lds(g0, g1, z4, z4, z8, 0);
}

// ---------------- prep kernels ----------------
__global__ void k_build_bet(const float* __restrict__ shape,
                            const float* __restrict__ expr,
                            float* __restrict__ BET) {
  int i = blockIdx.x * blockDim.x + threadIdx.x;
  if (i >= B_ * KP_) return;
  int b = i / KP_, l = i - b * KP_;
  float v = 0.f;
  if (l < 100)      v = shape[b * 100 + l];
  else if (l < 150) v = expr[b * 50 + (l - 100)];
  BET[i] = v;
}

__global__ void k_transpose_sd(const float* __restrict__ SD,
                               float* __restrict__ SDT) {
  int i = blockIdx.x * blockDim.x + threadIdx.x;
  if (i >= KP_ * NP_) return;
  int k = i / NP_, n = i - k * NP_;
  SDT[i] = (k < L_ && n < N_) ? SD[(size_t)n * L_ + k] : 0.f;
}

__global__ void k_pad_pd(const float* __restrict__ PD,
                         float* __restrict__ PDP) {
  int i = blockIdx.x * blockDim.x + threadIdx.x;
  if (i >= KPP_ * NP_) return;
  int k = i / NP_, n = i - k * NP_;
  PDP[i] = (k < 36 && n < N_) ? PD[(size_t)k * N_ + n] : 0.f;
}

__global__ void k_pad_vt(const float* __restrict__ vt, float* __restrict__ VT) {
  int n = blockIdx.x * blockDim.x + threadIdx.x;
  if (n >= NP_) return;
  VT[n] = (n < N_) ? vt[n] : 0.f;
}

// jdirs[jc][l] = sum_v Jreg[j,v]*shapedirs[v,c,l]; jtpl[jc] = sum_v Jreg[j,v]*vt[v,c]
__global__ void k_jdirs(const float* __restrict__ JR,
                        const float* __restrict__ SD,
                        const float* __restrict__ vt,
                        float* __restrict__ JD, float* __restrict__ JT) {
  int t = blockIdx.x * blockDim.x + threadIdx.x;
  if (t < 15 * KP_) {
    int jc = t / KP_, l = t - jc * KP_;
    float acc = 0.f;
    if (l < L_) {
      int j = jc / 3, c = jc - j * 3;
      const float* jr = JR + (size_t)j * V_;
      for (int v = 0; v < V_; ++v)
        acc += jr[v] * SD[((size_t)v * 3 + c) * L_ + l];
    }
    JD[t] = acc;
  } else if (t < 15 * KP_ + 15) {
    int jc = t - 15 * KP_;
    int j = jc / 3, c = jc - j * 3;
    const float* jr = JR + (size_t)j * V_;
    float acc = 0.f;
    for (int v = 0; v < V_; ++v) acc += jr[v] * vt[v * 3 + c];
    JT[jc] = acc;
  }
}

// ---------------- per-batch kernel ----------------
__device__ __forceinline__ void rodrigues3(const float* rv, float* R) {
  float a0 = rv[0] + 1e-8f, a1 = rv[1] + 1e-8f, a2 = rv[2] + 1e-8f;
  float ang = sqrtf(a0 * a0 + a1 * a1 + a2 * a2);
  float inv = 1.0f / ang;
  float nx = rv[0] * inv, ny = rv[1] * inv, nz = rv[2] * inv;
  float s = sinf(ang), c = cosf(ang), t = 1.0f - c;
  R[0] = 1.0f - t * (ny * ny + nz * nz);
  R[1] = -s * nz + t * nx * ny;
  R[2] =  s * ny + t * nx * nz;
  R[3] =  s * nz + t * nx * ny;
  R[4] = 1.0f - t * (nx * nx + nz * nz);
  R[5] = -s * nx + t * ny * nz;
  R[6] = -s * ny + t * nx * nz;
  R[7] =  s * nx + t * ny * nz;
  R[8] = 1.0f - t * (nx * nx + ny * ny);
}

// C = A*B for 3x4 affine (implicit bottom row [0,0,0,1])
__device__ __forceinline__ void amul(const float* A, const float* Bm, float* C) {
  for (int r = 0; r < 3; ++r)
    for (int c = 0; c < 4; ++c) {
      float s = A[r * 4 + 0] * Bm[0 * 4 + c] + A[r * 4 + 1] * Bm[1 * 4 + c] +
                A[r * 4 + 2] * Bm[2 * 4 + c];
      if (c == 3) s += A[r * 4 + 3];
      C[r * 4 + c] = s;
    }
}

__global__ __launch_bounds__(256) void k_batch(
    const float* __restrict__ pose, const float* __restrict__ neck,
    const float* __restrict__ eye, const float* __restrict__ BET,
    const float* __restrict__ JD, const float* __restrict__ JT,
    float* __restrict__ PFP, float* __restrict__ RTF, int* __restrict__ YIDX) {
  int b = blockIdx.x * blockDim.x + threadIdx.x;
  if (b >= B_) return;
  float fp[15];
  fp[0] = pose[b * 6 + 0]; fp[1] = pose[b * 6 + 1]; fp[2] = pose[b * 6 + 2];
  fp[3] = neck[0]; fp[4] = neck[1]; fp[5] = neck[2];
  fp[6] = pose[b * 6 + 3]; fp[7] = pose[b * 6 + 4]; fp[8] = pose[b * 6 + 5];
  fp[9]  = eye[0]; fp[10] = eye[1]; fp[11] = eye[2];
  fp[12] = eye[3]; fp[13] = eye[4]; fp[14] = eye[5];

  float R[5][9];
  for (int j = 0; j < 5; ++j) rodrigues3(&fp[3 * j], R[j]);

  // pose_feature = (R[1:] - I).flat, padded to 40
  for (int j = 1; j < 5; ++j)
    for (int e = 0; e < 9; ++e)
      PFP[(size_t)b * KPP_ + (j - 1) * 9 + e] =
          R[j][e] - ((e == 0 || e == 4 || e == 8) ? 1.f : 0.f);
  for (int e = 36; e < KPP_; ++e) PFP[(size_t)b * KPP_ + e] = 0.f;

  // joints[jc] = jtpl + jdirs . betas
  float jac[15];
  for (int t = 0; t < 15; ++t) jac[t] = JT[t];
  for (int l = 0; l < L_; ++l) {
    float bl = BET[(size_t)b * KP_ + l];
    for (int t = 0; t < 15; ++t) jac[t] += bl * JD[t * KP_ + l];
  }

  const int par[5] = {-1, 0, 1, 1, 1};
  float tm[5][12];
  for (int j = 0; j < 5; ++j) {
    float rj[3];
    for (int c = 0; c < 3; ++c)
      rj[c] = jac[j * 3 + c] - (j > 0 ? jac[par[j] * 3 + c] : 0.f);
    for (int r = 0; r < 3; ++r) {
      tm[j][r * 4 + 0] = R[j][r * 3 + 0];
      tm[j][r * 4 + 1] = R[j][r * 3 + 1];
      tm[j][r * 4 + 2] = R[j][r * 3 + 2];
      tm[j][r * 4 + 3] = rj[r];
    }
  }
  float Tr[5][12];
  for (int e = 0; e < 12; ++e) Tr[0][e] = tm[0][e];
  amul(Tr[0], tm[1], Tr[1]);
  amul(Tr[1], tm[2], Tr[2]);
  amul(Tr[1], tm[3], Tr[3]);
  amul(Tr[1], tm[4], Tr[4]);

  // rel_tf: subtract (R_part * joint) from translation column
  for (int j = 0; j < 5; ++j) {
    float jx = jac[j * 3 + 0], jy = jac[j * 3 + 1], jz = jac[j * 3 + 2];
    for (int r = 0; r < 3; ++r) {
      float t3 = Tr[j][r * 4 + 3] -
                 (Tr[j][r * 4 + 0] * jx + Tr[j][r * 4 + 1] * jy + Tr[j][r * 4 + 2] * jz);
      RTF[(size_t)b * 60 + j * 12 + r * 4 + 0] = Tr[j][r * 4 + 0];
      RTF[(size_t)b * 60 + j * 12 + r * 4 + 1] = Tr[j][r * 4 + 1];
      RTF[(size_t)b * 60 + j * 12 + r * 4 + 2] = Tr[j][r * 4 + 2];
      RTF[(size_t)b * 60 + j * 12 + r * 4 + 3] = t3;
    }
  }

  // dynamic landmark bin: rel_rot = R[0] @ R[1]; only column 0 needed
  float r00 = R[0][0] * R[1][0] + R[0][1] * R[1][3] + R[0][2] * R[1][6];
  float r10 = R[0][3] * R[1][0] + R[0][4] * R[1][3] + R[0][5] * R[1][6];
  float r20 = R[0][6] * R[1][0] + R[0][7] * R[1][3] + R[0][8] * R[1][6];
  float sy = sqrtf(r00 * r00 + r10 * r10);
  float yang = atan2f(-r20, sy) * 57.29577951308232f;
  int y = (int)rintf(fminf(yang, 39.0f));
  if (y < 0) y = (y < -39) ? 78 : 39 - y;
  YIDX[b] = y;
}

// ---------------- fused shape+pose blend GEMM (WMMA f32 16x16x4) ----------------
// v_posed[b,n] = VT[n] + sum_l BET[b,l]*SDT[l,n] + sum_p PFP[b,p]*PDP[p,n]
__global__ __launch_bounds__(256) void k_gemm(
    const float* __restrict__ BET, const float* __restrict__ PFP,
    const float* __restrict__ SDT, const float* __restrict__ PDP,
    const float* __restrict__ VT, float* __restrict__ VP) {
  __shared__ __align__(16) float bet_s[16 * KP_];
  __shared__ __align__(16) float pf_s[16 * KPP_];
  const int b0 = blockIdx.y << 4;
  const int wave = threadIdx.x >> 5;
  const int lane = threadIdx.x & 31;

  if (wave == 0) {  // TDM ignores EXEC but issues per-wave: issue from wave 0 only
    tdm_load_2d((u32)(unsigned long long)(void*)bet_s,
                BET + (size_t)b0 * KP_, KP_, 16);
    tdm_load_2d((u32)(unsigned long long)(void*)pf_s,
                PFP + (size_t)b0 * KPP_, KPP_, 16);
    __builtin_amdgcn_s_wait_tensorcnt(0);
  }
  __syncthreads();

  const int nl = lane & 15;                    // B/C/D column
  const int n  = (blockIdx.x << 7) + (wave << 4) + nl;
  const int kh = (lane >> 4) << 1;             // K sub-pair: 0 or 2
  const int m  = lane & 15;                    // A row

  v8f acc = {};
  #pragma unroll 4
  for (int k0 = 0; k0 < KP_; k0 += 4) {
    v2f a = *(const v2f*)&bet_s[m * KP_ + k0 + kh];     // A: K=k0+kh, k0+kh+1
    v2f bb;
    bb.x = SDT[(size_t)(k0 + kh) * NP_ + n];            // B rows k0+kh (VGPR0)
    bb.y = SDT[(size_t)(k0 + kh + 1) * NP_ + n];        // B rows k0+kh+1 (VGPR1)
    if (k0 + 8 < KP_)
      __builtin_prefetch(&SDT[(size_t)(k0 + 8 + kh) * NP_ + n], 0, 1);
    acc = __builtin_amdgcn_wmma_f32_16x16x4_f32(false, a, false, bb,
                                                (short)0, acc, false, false);
  }
  #pragma unroll
  for (int k0 = 0; k0 < KPP_; k0 += 4) {
    v2f a = *(const v2f*)&pf_s[m * KPP_ + k0 + kh];
    v2f bb;
    bb.x = PDP[(size_t)(k0 + kh) * NP_ + n];
    bb.y = PDP[(size_t)(k0 + kh + 1) * NP_ + n];
    acc = __builtin_amdgcn_wmma_f32_16x16x4_f32(false, a, false, bb,
                                                (short)0, acc, false, false);
  }

  const float vt = VT[n];
  #pragma unroll
  for (int r = 0; r < 8; ++r) {
    const int mr = ((lane >> 4) << 3) + r;     // C/D row map
    VP[(size_t)(b0 + mr) * NP_ + n] = acc[r] + vt;
  }
}

// ---------------- LBS skinning ----------------
__global__ __launch_bounds__(128) void k_lbs(
    const float* __restrict__ VP, const float* __restrict__ RTF,
    const float* __restrict__ W, float* __restrict__ out) {
  const int b = blockIdx.y;
  __shared__ float rt[60];
  if (threadIdx.x < 60) rt[threadIdx.x] = RTF[(size_t)b * 60 + threadIdx.x];
  __syncthreads();
  const int v = blockIdx.x * 128 + threadIdx.x;
  if (v >= V_) return;
  float w[5];
  for (int j = 0; j < 5; ++j) w[j] = W[(size_t)v * 5 + j];
  const float* vp = &VP[(size_t)b * NP_ + (size_t)v * 3];
  const float x = vp[0], y = vp[1], z = vp[2];
  for (int c = 0; c < 3; ++c) {
    float s = 0.f;
    for (int j = 0; j < 5; ++j) {
      const float* M = &rt[j * 12 + c * 4];
      s += w[j] * (M[0] * x + M[1] * y + M[2] * z + M[3]);
    }
    out[(size_t)b * (V_ * 3) + (size_t)v * 3 + c] = s;
  }
}

// ---------------- landmarks ----------------
__global__ __launch_bounds__(416) void k_lmk(
    const float* __restrict__ verts, const int* __restrict__ yidx,
    const int* __restrict__ faces, const int* __restrict__ lmkF,
    const float* __restrict__ lmkB, const int* __restrict__ dynF,
    const float* __restrict__ dynB, const int* __restrict__ fullF,
    const float* __restrict__ fullB, float* __restrict__ out2d,
    float* __restrict__ out3d) {
  const int b = blockIdx.x;
  const int t = threadIdx.x;
  if (t >= 2 * NLMK_ * 3) return;
  const int which = (t >= NLMK_ * 3) ? 1 : 0;    // 0 -> 2d, 1 -> 3d
  const int r = t - which * (NLMK_ * 3);
  const int l = r / 3, c = r - l * 3;
  int f;
  float b0, b1, b2;
  if (!which) {
    if (l < 17) {
      const int y = yidx[b];
      const int idx = y * 17 + l;
      f = dynF[idx];
      b0 = dynB[idx * 3 + 0]; b1 = dynB[idx * 3 + 1]; b2 = dynB[idx * 3 + 2];
    } else {
      const int ll = l - 17;
      f = lmkF[ll];
      b0 = lmkB[ll * 3 + 0]; b1 = lmkB[ll * 3 + 1]; b2 = lmkB[ll * 3 + 2];
    }
  } else {
    f = fullF[l];
    b0 = fullB[l * 3 + 0]; b1 = fullB[l * 3 + 1]; b2 = fullB[l * 3 + 2];
  }
  const int t0 = faces[f * 3 + 0], t1 = faces[f * 3 + 1], t2 = faces[f * 3 + 2];
  const size_t base = (size_t)b * (V_ * 3);
  const float val = b0 * verts[base + (size_t)t0 * 3 + c] +
                    b1 * verts[base + (size_t)t1 * 3 + c] +
                    b2 * verts[base + (size_t)t2 * 3 + c];
  if (!which) out2d[((size_t)b * NLMK_ + l) * 3 + c] = val;
  else        out3d[((size_t)b * NLMK_ + l) * 3 + c] = val;
}

// ---------------- host launcher ----------------
extern "C" void kernel_launch(void* const* d_in, const int* in_sizes, int n_in,
                              void* d_out, int out_size, void* d_ws, size_t ws_size,
                              hipStream_t stream) {
  const float* shape = (const float*)d_in[0];
  const float* expr  = (const float*)d_in[1];
  const float* pose  = (const float*)d_in[2];
  const float* vtmpl = (const float*)d_in[3];
  const float* sd    = (const float*)d_in[4];
  const float* pd    = (const float*)d_in[5];
  const float* jreg  = (const float*)d_in[6];
  const float* lbsw  = (const float*)d_in[7];
  const float* neck  = (const float*)d_in[8];
  const float* eye   = (const float*)d_in[9];
  const int*   faces = (const int*)d_in[10];
  const int*   lmkF  = (const int*)d_in[11];
  const float* lmkB  = (const float*)d_in[12];
  const int*   dynF  = (const int*)d_in[13];
  const float* dynB  = (const float*)d_in[14];
  const int*   fullF = (const int*)d_in[15];
  const float* fullB = (const float*)d_in[16];

  char* ws = (char*)d_ws;
  size_t off = 0;
  auto alloc = [&](size_t bytes) {
    char* p = ws + off;
    off = (off + bytes + 255) & ~(size_t)255;
    return p;
  };
  float* BET  = (float*)alloc((size_t)B_ * KP_ * 4);
  float* PFP  = (float*)alloc((size_t)B_ * KPP_ * 4);
  float* SDT  = (float*)alloc((size_t)KP_ * NP_ * 4);
  float* PDP  = (float*)alloc((size_t)KPP_ * NP_ * 4);
  float* VT   = (float*)alloc((size_t)NP_ * 4);
  float* JD   = (float*)alloc((size_t)15 * KP_ * 4);
  float* JT   = (float*)alloc(64);
  float* RTF  = (float*)alloc((size_t)B_ * 60 * 4);
  int*   YIDX = (int*)alloc((size_t)B_ * 4);
  float* VP   = (float*)alloc((size_t)B_ * NP_ * 4);

  float* Overt = (float*)d_out;
  float* O2d   = Overt + (size_t)B_ * V_ * 3;
  float* O3d   = O2d + (size_t)B_ * NLMK_ * 3;

  k_build_bet<<<(B_ * KP_ + 255) / 256, 256, 0, stream>>>(shape, expr, BET);
  k_transpose_sd<<<(KP_ * NP_ + 255) / 256, 256, 0, stream>>>(sd, SDT);
  k_pad_pd<<<(KPP_ * NP_ + 255) / 256, 256, 0, stream>>>(pd, PDP);
  k_pad_vt<<<(NP_ + 255) / 256, 256, 0, stream>>>(vtmpl, VT);
  k_jdirs<<<(15 * KP_ + 15 + 255) / 256, 256, 0, stream>>>(jreg, sd, vtmpl, JD, JT);
  k_batch<<<B_ / 256, 256, 0, stream>>>(pose, neck, eye, BET, JD, JT, PFP, RTF, YIDX);
  k_gemm<<<dim3(NP_ / 128, B_ / 16), 256, 0, stream>>>(BET, PFP, SDT, PDP, VT, VP);
  k_lbs<<<dim3((V_ + 127) / 128, B_), 128, 0, stream>>>(VP, RTF, lbsw, Overt);
  k_lmk<<<B_, 416, 0, stream>>>(Overt, YIDX, faces, lmkF, lmkB, dynF, dynB,
                                fullF, fullB, O2d, O3d);
  (void)in_sizes; (void)n_in; (void)out_size; (void)ws_size;
}